// QwenLinearAttentionBlock_62612033241794
// MI455X (gfx1250) — compile-verified
//
#include <hip/hip_runtime.h>
#include <hip/hip_bf16.h>
#include <math.h>

#define HH   32
#define DD   128
#define HID  4096
#define II   (HH * DD)          // 4096
#define EPSF 1e-6f

typedef __attribute__((ext_vector_type(2))) float v2f;
typedef __attribute__((ext_vector_type(8))) float v8f;

// gfx1250 async global->LDS copy path (ASYNCcnt-tracked), if the toolchain
// exposes the builtins; otherwise fall back to a synchronous float4 copy.
#if __has_builtin(__builtin_amdgcn_global_load_async_to_lds_b128) && \
    __has_builtin(__builtin_amdgcn_s_wait_asynccnt)
#define HAVE_ASYNC_LDS 1
#endif

#ifdef HAVE_ASYNC_LDS
// exact pointee type per clang diagnostic: int __attribute__((vector_size(16)))
typedef int b128_t __attribute__((vector_size(16)));
typedef __attribute__((address_space(1))) b128_t* g_b128_p;  // global
typedef __attribute__((address_space(3))) b128_t* l_b128_p;  // LDS
#endif

__device__ __forceinline__ void copy16_to_lds(const float* g, float* l) {
#ifdef HAVE_ASYNC_LDS
  __builtin_amdgcn_global_load_async_to_lds_b128(
      (g_b128_p)(size_t)g, (l_b128_p)(b128_t*)l, /*offset=*/0, /*cpol=*/0);
#else
  *(float4*)l = *(const float4*)g;
#endif
}

__device__ __forceinline__ void wait_lds_copies() {
#ifdef HAVE_ASYNC_LDS
  __builtin_amdgcn_s_wait_asynccnt(0);
#endif
}

// ---------------------------------------------------------------------------
// GEMV  y = concat(W0,W1,W2,W3) @ x   via V_WMMA_F32_16X16X4_F32
// Block = 256 thr (8 waves); each wave owns 16 output rows; 128 rows / block.
// Double-buffered HBM->LDS tiles (async when available), WMMA consumes LDS.
// B = x broadcast to all 16 N columns (layout-robust; FLOPs are free here).
// ---------------------------------------------------------------------------
#define KC 32
#define TSTRIDE 36   // dwords; 36*4B rows are 16B aligned; 36*dr mod 64 never
                     // within +-3 of 0 for dr=1..15 -> wmma LDS reads are
                     // bank-conflict-free for the 16-row x 4-col pattern

__device__ __forceinline__ const float* seg_row(
    const float* __restrict__ W0, int n0, const float* __restrict__ W1, int n1,
    const float* __restrict__ W2, int n2, const float* __restrict__ W3, int n3,
    int row) {
  if (row < n0) return W0 + (size_t)row * HID;
  row -= n0; if (row < n1) return W1 + (size_t)row * HID;
  row -= n1; if (row < n2) return W2 + (size_t)row * HID;
  row -= n2; if (row < n3) return W3 + (size_t)row * HID;
  return W0;  // clamp (guarded rows: loads valid memory, store suppressed)
}

__global__ __launch_bounds__(256) void gemv_wmma_kernel(
    const float* __restrict__ W0, int n0,
    const float* __restrict__ W1, int n1,
    const float* __restrict__ W2, int n2,
    const float* __restrict__ W3, int n3,
    const float* __restrict__ x,
    float* __restrict__ y) {
  __shared__ __align__(16) float xs[HID];
  __shared__ __align__(16) float tile[2][128 * TSTRIDE];

  const int nrows    = n0 + n1 + n2 + n3;
  const int blockRow = blockIdx.x * 128;
  const int tid      = threadIdx.x;
  const int lane     = tid & 31;
  const int wave     = tid >> 5;

  // whole x vector into LDS once (16 KB), vectorized + coalesced
  for (int i = tid * 4; i < HID; i += 256 * 4)
    *(float4*)&xs[i] = *(const float4*)&x[i];

  // A-matrix read coords per the 32-bit 16x4 WMMA layout:
  // lane L holds A[L%16][2*(L/16) + vgpr]
  const int kHalf = (lane >> 4) * 2;          // 0 or 2
  const int arow  = wave * 16 + (lane & 15);  // tile row for this lane

  // tile staging map: 8 consecutive threads cover one row's 32 floats (128 B)
  const int trow = tid >> 3;       // 0..31, stepped by 32 (4 iterations)
  const int tcol = (tid & 7) * 4;  // dword col (float4 granule)

  v8f acc = {0.f, 0.f, 0.f, 0.f, 0.f, 0.f, 0.f, 0.f};

  // prologue: stage chunk 0 into buffer 0
  for (int r = trow; r < 128; r += 32) {
    const float* src = seg_row(W0, n0, W1, n1, W2, n2, W3, n3, blockRow + r);
    copy16_to_lds(src + tcol, &tile[0][r * TSTRIDE + tcol]);
  }
  wait_lds_copies();
  __syncthreads();

  for (int c = 0; c < HID / KC; ++c) {
    const int cur = c & 1;
    const int k0  = c * KC;

    // issue async staging of the next chunk into the other buffer
    if (k0 + KC < HID) {
      for (int r = trow; r < 128; r += 32) {
        const float* src = seg_row(W0, n0, W1, n1, W2, n2, W3, n3, blockRow + r);
        copy16_to_lds(src + k0 + KC + tcol, &tile[1 - cur][r * TSTRIDE + tcol]);
      }
    }

#if __has_builtin(__builtin_amdgcn_wmma_f32_16x16x4_f32)
#pragma unroll
    for (int s = 0; s < KC / 4; ++s) {
      v2f a = *(const v2f*)(&tile[cur][arow * TSTRIDE + s * 4 + kHalf]);
      v2f b = *(const v2f*)(xs + k0 + s * 4 + kHalf);  // broadcast to all N
      acc = __builtin_amdgcn_wmma_f32_16x16x4_f32(
          /*neg_a=*/false, a, /*neg_b=*/false, b,
          /*c_mod=*/(short)0, acc, /*reuse_a=*/false, /*reuse_b=*/false);
    }
#else
    // scalar fallback (compile-robustness only): lanes 0/16 own 8 rows each
    if ((lane & 15) == 0) {
      int rb = wave * 16 + (lane >> 4) * 8;
      for (int s = 0; s < KC / 4; ++s)
        for (int i = 0; i < 8; ++i)
          for (int j = 0; j < 4; ++j)
            acc[i] += tile[cur][(rb + i) * TSTRIDE + s * 4 + j] *
                      xs[k0 + s * 4 + j];
    }
#endif

    wait_lds_copies();   // my async loads into the other buffer are done
    __syncthreads();     // everyone finished reading cur + staging other
  }

  // D layout: lanes 0-15 VGPR m = row m ; lanes 16-31 VGPR m = row m+8.
  // All N columns identical -> use lane 0 (rows 0-7) and lane 16 (rows 8-15).
  if ((lane & 15) == 0) {
    int rbase = blockRow + wave * 16 + (lane >> 4) * 8;
#pragma unroll
    for (int i = 0; i < 8; ++i) {
      int r = rbase + i;
      if (r < nrows) y[r] = acc[i];
    }
  }
}

// ---------------------------------------------------------------------------
// Per-head: conv(4-tap)+silu, gates, L2 norms, gated delta-rule state update,
// RMS norm, core * silu(z).  One block per head, 256 threads.
// ---------------------------------------------------------------------------
__global__ __launch_bounds__(256) void head_update_kernel(
    const float* __restrict__ rnn_state,   // [H][D][D]
    const float* __restrict__ conv_state,  // [3I][3]
    const float* __restrict__ conv_w,      // [3I][4]
    const float* __restrict__ conv_b,      // [3I]
    const float* __restrict__ A_log,       // [H]
    const float* __restrict__ dt_bias,     // [H]
    const float* __restrict__ norm_w,      // [D]
    const float* __restrict__ yall,        // ws: qkv[3I] z[I] a[H] b[H]
    float* __restrict__ core_z,            // ws out [I]
    float* __restrict__ state_out,         // d_out + HID
    float* __restrict__ conv_out) {        // d_out + HID + H*D*D
  __shared__ float q[DD], k[DD], v[DD];
  __shared__ float red[2 * DD];
  __shared__ float scal[4];

  const int h   = blockIdx.x;
  const int tid = threadIdx.x;
  const float* qkv  = yall;
  const float* z    = yall + 3 * II;
  const float* aarr = yall + 3 * II + II;       // a projections
  const float* barr = aarr + HH;                // b projections

  // conv over this head's 3*D channels; also emit new_conv_state
  for (int cl = tid; cl < 3 * DD; cl += 256) {
    int seg = cl / DD, d = cl % DD;
    int ch = seg * II + h * DD + d;
    float c0 = conv_state[ch * 3 + 0];
    float c1 = conv_state[ch * 3 + 1];
    float c2 = conv_state[ch * 3 + 2];
    float xin = qkv[ch];
    float yv = c0 * conv_w[ch * 4 + 0] + c1 * conv_w[ch * 4 + 1] +
               c2 * conv_w[ch * 4 + 2] + xin * conv_w[ch * 4 + 3] + conv_b[ch];
    yv = yv / (1.f + expf(-yv));  // silu
    if (seg == 0) q[d] = yv; else if (seg == 1) k[d] = yv; else v[d] = yv;
    conv_out[ch * 3 + 0] = c1;
    conv_out[ch * 3 + 1] = c2;
    conv_out[ch * 3 + 2] = xin;
  }
  __syncthreads();

  // L2-norm partials
  if (tid < DD) { red[tid] = q[tid] * q[tid]; red[DD + tid] = k[tid] * k[tid]; }
  __syncthreads();
  for (int s2 = DD / 2; s2 > 0; s2 >>= 1) {
    if (tid < s2) { red[tid] += red[tid + s2]; red[DD + tid] += red[DD + tid + s2]; }
    __syncthreads();
  }
  if (tid == 0) {
    float ah = aarr[h] + dt_bias[h];
    float sp = (ah > 20.f) ? ah : log1pf(expf(ah));   // softplus
    scal[0] = expf(-expf(A_log[h]) * sp);             // exp(g)
    scal[1] = 1.f / (1.f + expf(-barr[h]));           // beta
    scal[2] = 1.f / fmaxf(sqrtf(red[0]), EPSF);       // 1/||q||
    scal[3] = 1.f / fmaxf(sqrtf(red[DD]), EPSF);      // 1/||k||
  }
  __syncthreads();
  const float eg   = scal[0];
  const float beta = scal[1];
  const float invq = scal[2] * 0.08838834764831845f;  // * 1/sqrt(128)
  const float invk = scal[3];

  const int e    = tid & (DD - 1);
  const int half = tid >> 7;          // 0/1: d-range split
  const int dlo  = half * (DD / 2);
  const float* srow = rnn_state + (size_t)h * DD * DD;
  __syncthreads();   // red about to be reused

  // pass 1: kv_mem[e] = sum_d eg*s[d][e]*kn[d]
  float kv = 0.f;
  for (int d = dlo; d < dlo + DD / 2; ++d)
    kv += (eg * srow[d * DD + e]) * (k[d] * invk);
  red[tid] = kv;
  __syncthreads();
  const float delta = (v[e] - (red[e] + red[DD + e])) * beta;
  __syncthreads();

  // pass 2: s' = eg*s + kn*delta ; core[e] = sum_d s'[d][e]*qn[d]
  float core = 0.f;
  float* sout = state_out + (size_t)h * DD * DD;
  for (int d = dlo; d < dlo + DD / 2; ++d) {
    float sv = eg * srow[d * DD + e] + (k[d] * invk) * delta;
    sout[d * DD + e] = sv;
    core += sv * (q[d] * invq);
  }
  red[tid] = core;
  __syncthreads();
  const float core_e = red[e] + red[DD + e];
  __syncthreads();

  // RMS norm over D
  if (tid < DD) red[tid] = core_e * core_e;
  __syncthreads();
  for (int s2 = DD / 2; s2 > 0; s2 >>= 1) {
    if (tid < s2) red[tid] += red[tid + s2];
    __syncthreads();
  }
  const float rn = rsqrtf(red[0] * (1.f / DD) + EPSF);
  if (tid < DD) {
    float cn = norm_w[e] * core_e * rn;
    float zv = z[h * DD + e];
    core_z[h * DD + e] = cn * (zv / (1.f + expf(-zv)));
  }
}

// ---------------------------------------------------------------------------
extern "C" void kernel_launch(void* const* d_in, const int* in_sizes, int n_in,
                              void* d_out, int out_size, void* d_ws, size_t ws_size,
                              hipStream_t stream) {
  (void)in_sizes; (void)n_in; (void)out_size; (void)ws_size;
  const float* hidden  = (const float*)d_in[0];
  const float* rnn     = (const float*)d_in[1];
  const float* convs   = (const float*)d_in[2];
  const float* W_qkv   = (const float*)d_in[3];
  const float* W_z     = (const float*)d_in[4];
  const float* W_a     = (const float*)d_in[5];
  const float* W_b     = (const float*)d_in[6];
  const float* conv_w  = (const float*)d_in[7];
  const float* conv_b  = (const float*)d_in[8];
  const float* A_log   = (const float*)d_in[9];
  const float* dt_bias = (const float*)d_in[10];
  const float* norm_w  = (const float*)d_in[11];
  const float* W_out   = (const float*)d_in[12];

  float* ws    = (float*)d_ws;
  float* yall  = ws;            // 16448 = qkv(12288) z(4096) a(32) b(32)
  float* corez = ws + 16448;    // 4096

  float* out       = (float*)d_out;            // [4096]
  float* state_out = out + HID;                // [32*128*128]
  float* conv_out  = state_out + HH * DD * DD; // [12288*3]

  // phase 1: fused qkv/z/a/b projections (16448 rows -> 129 blocks)
  gemv_wmma_kernel<<<129, 256, 0, stream>>>(W_qkv, 3 * II, W_z, II,
                                            W_a, HH, W_b, HH, hidden, yall);
  // phase 2: conv + gates + state update + RMS + core*silu(z)
  head_update_kernel<<<HH, 256, 0, stream>>>(rnn, convs, conv_w, conv_b, A_log,
                                             dt_bias, norm_w, yall, corez,
                                             state_out, conv_out);
  // phase 3: out = W_out @ corez (4096 rows -> 32 blocks)
  gemv_wmma_kernel<<<32, 256, 0, stream>>>(W_out, HID, W_out, 0,
                                           W_out, 0, W_out, 0, corez, out);
}